// GraphConvolution_18597208391760
// MI455X (gfx1250) — compile-verified
//
#include <hip/hip_runtime.h>

#define BB 2
#define RR 2
#define NN 50000
#define EE 500000
#define DIN 128
#define DOUT 128

typedef __attribute__((ext_vector_type(2))) float v2f;
typedef __attribute__((ext_vector_type(8))) float v8f;

// ---------------------------------------------------------------------------
// Kernel 1: pre[b,r,n,:] = x[b,n,:] @ W[r]   via V_WMMA_F32_16X16X4_F32
// Block = 256 threads (8 waves). Each wave computes a 16-row x 128-col tile.
// W[r] (64KB) staged in LDS; B-fragments via ds_load_2addr_b32.
// A-fragment global loads are software-pipelined one k-step ahead (the k-loop
// is fully unrolled so the pipeline guard constant-folds), hiding global
// latency behind the previous step's 8 WMMAs.
// A-fragment layout (16x4 f32): lanes 0-15 rows M=0..15 K=0,1 ; lanes 16-31
// same rows K=2,3. C/D layout: vgpr v -> M=v (lanes 0-15), M=v+8 (lanes
// 16-31), N = lane&15.
// ---------------------------------------------------------------------------
__global__ __launch_bounds__(256) void rgcn_gemm_wmma(const float* __restrict__ x,
                                                      const float* __restrict__ w,
                                                      float* __restrict__ pre) {
  __shared__ float wlds[DIN * DOUT]; // 64 KB

  const int wave = threadIdx.x >> 5;
  const int lane = threadIdx.x & 31;

  const int tiles = NN / 16;                 // 3125 row tiles (exact)
  const int tpb   = 8;                       // tiles per block (8 waves)
  const int bpr   = (tiles + tpb - 1) / tpb; // 391 blocks per (b,r)
  const int br    = blockIdx.x / bpr;        // 0..3 => b*R + r
  const int tb    = blockIdx.x % bpr;
  const int r     = br % RR;
  const int b     = br / RR;

  // Stage W[r] into LDS (128x128 f32), 16B vector copies.
  const float* wr = w + (size_t)r * DIN * DOUT;
  for (int i = threadIdx.x * 4; i < DIN * DOUT; i += 256 * 4) {
    *(float4*)&wlds[i] = *(const float4*)&wr[i];
  }
  __syncthreads();

  const int tile = tb * tpb + wave;
  if (tile >= tiles) return;                 // wave-uniform exit: EXEC stays all-1s
  const int row0 = tile * 16;

  const int half = lane >> 4;                // 0: K=0,1  1: K=2,3
  const int l    = lane & 15;
  const float* xrow = x + ((size_t)b * NN + row0 + l) * DIN + half * 2;

  v8f acc[8] = {};

  // Prime the A-fragment pipeline.
  v2f a_cur;
  a_cur.x = xrow[0];
  a_cur.y = xrow[1];

#pragma unroll
  for (int k = 0; k < DIN; k += 4) {
    v2f a_nxt = a_cur;
    if (k + 4 < DIN) {                       // constant-folded under full unroll
      a_nxt.x = xrow[k + 4 + 0];
      a_nxt.y = xrow[k + 4 + 1];
    }
#pragma unroll
    for (int ct = 0; ct < 8; ++ct) {
      const float* wp = &wlds[(k + half * 2) * DOUT + ct * 16 + l];
      v2f bf;
      bf.x = wp[0];
      bf.y = wp[DOUT];
      acc[ct] = __builtin_amdgcn_wmma_f32_16x16x4_f32(
          /*neg_a=*/false, a_cur, /*neg_b=*/false, bf,
          /*c_mod=*/(short)0, acc[ct], /*reuse_a=*/false, /*reuse_b=*/false);
    }
    a_cur = a_nxt;
  }

  // Store the 16x128 tile.
  float* prebase = pre + (((size_t)br) * NN + row0) * DOUT;
#pragma unroll
  for (int ct = 0; ct < 8; ++ct) {
#pragma unroll
    for (int v = 0; v < 8; ++v) {
      const int m = v + half * 8;
      prebase[(size_t)m * DOUT + ct * 16 + l] = acc[ct][v];
    }
  }
}

// ---------------------------------------------------------------------------
// Kernel 2: edge scatter. One wave per (b,r,e): gather pre[b,r,src,:] (512B,
// fully coalesced b128 loads), scale by edge_val, atomicAdd into out[b,dst,:].
// Sum over relations happens implicitly in the atomics.
// ---------------------------------------------------------------------------
__global__ __launch_bounds__(256) void rgcn_scatter(const float* __restrict__ pre,
                                                    const float* __restrict__ edge_val,
                                                    const int* __restrict__ edge_src,
                                                    const int* __restrict__ edge_dst,
                                                    float* __restrict__ out) {
  const long long gid  = (long long)blockIdx.x * blockDim.x + threadIdx.x;
  const long long eidx = gid >> 5;                 // edge over B*R*E
  const int lane = threadIdx.x & 31;
  if (eidx >= (long long)BB * RR * EE) return;

  const int br = (int)(eidx / EE);                 // b*R + r
  const int b  = br / RR;

  const int   src = edge_src[eidx];
  const int   dst = edge_dst[eidx];
  const float val = edge_val[eidx];

  const float* p = pre + ((size_t)br * NN + src) * DOUT + lane * 4;
  float*       o = out + ((size_t)b  * NN + dst) * DOUT + lane * 4;

  float4 m = *(const float4*)p;                    // global_load_b128, coalesced
  atomicAdd(o + 0, m.x * val);
  atomicAdd(o + 1, m.y * val);
  atomicAdd(o + 2, m.z * val);
  atomicAdd(o + 3, m.w * val);
}

// ---------------------------------------------------------------------------
// Kernel 3: out = relu(out + bias), in place.
// ---------------------------------------------------------------------------
__global__ __launch_bounds__(256) void rgcn_bias_relu(float* __restrict__ out,
                                                      const float* __restrict__ bias) {
  const size_t i = (size_t)blockIdx.x * blockDim.x + threadIdx.x;
  if (i >= (size_t)BB * NN * DOUT) return;
  const float v = out[i] + bias[i & (DOUT - 1)];
  out[i] = v > 0.f ? v : 0.f;
}

extern "C" void kernel_launch(void* const* d_in, const int* in_sizes, int n_in,
                              void* d_out, int out_size, void* d_ws, size_t ws_size,
                              hipStream_t stream) {
  const float* x        = (const float*)d_in[0];
  const float* edge_val = (const float*)d_in[1];
  const float* w        = (const float*)d_in[2];
  const float* bias     = (const float*)d_in[3];
  const int*   edge_src = (const int*)d_in[4];
  const int*   edge_dst = (const int*)d_in[5];
  float* out = (float*)d_out;
  float* pre = (float*)d_ws;   // B*R*N*DOUT floats = 102.4 MB scratch

  // Atomics accumulate into d_out -> must zero it each call (capture-safe).
  hipMemsetAsync(d_out, 0, (size_t)out_size * sizeof(float), stream);

  const int tiles = NN / 16;
  const int bpr   = (tiles + 7) / 8;
  rgcn_gemm_wmma<<<BB * RR * bpr, 256, 0, stream>>>(x, w, pre);

  const long long total_threads = (long long)BB * RR * EE * 32;
  rgcn_scatter<<<(int)((total_threads + 255) / 256), 256, 0, stream>>>(
      pre, edge_val, edge_src, edge_dst, out);

  const size_t tot = (size_t)BB * NN * DOUT;
  rgcn_bias_relu<<<(int)((tot + 255) / 256), 256, 0, stream>>>(out, bias);
}